// DCMP_block_19181323944523
// MI455X (gfx1250) — compile-verified
//
#include <hip/hip_runtime.h>
#include <stdint.h>

// Problem constants (fixed by setup_inputs: x[32,336,32,32] f32, k=25, pe=24)
#define B_    32
#define C_    336
#define HW_   1024
#define PE_   24
#define TILE_ 64
#define NTHR_ 64
#define OFFS_ ((size_t)B_ * C_ * HW_)   // 11,010,048 floats per output tensor

typedef unsigned int u32x4 __attribute__((ext_vector_type(4)));
typedef unsigned int u32x8 __attribute__((ext_vector_type(8)));

__global__ __launch_bounds__(NTHR_) void decomp_tdm_kernel(const float* __restrict__ x,
                                                           float* __restrict__ out)
{
    // [C_][TILE_] tile of x: 336 * 64 * 4B = 86016 B of LDS
    __shared__ float tile[C_ * TILE_];

    const int tid = threadIdx.x;        // 0..63
    const int wid = tid >> 5;           // wave id: 0 or 1 (uniform per wave)
    const int blk = blockIdx.x;
    const int b   = blk >> 4;           // HW_/TILE_ = 16 tiles per batch
    const int t0  = (blk & 15) * TILE_;

    const float* src = x + (size_t)b * (C_ * HW_) + t0;   // channel c row at src + c*HW_

    // ---------------- TDM: DMA the strided 2D tile global -> LDS ----------------
    // Each wave moves half the channel rows (168 rows x 64 floats, pitch 1024 floats).
    {
        const int rows = C_ / 2;                    // 168
        const int r0   = wid * rows;
        const uint64_t ga  = (uint64_t)(uintptr_t)(src + (size_t)r0 * HW_);
        const uint32_t lds = ((uint32_t)(uintptr_t)(&tile[0])) + (uint32_t)(r0 * TILE_ * 4);

        // D# group 0 (128b): count=1 | lds_addr | global_addr[56:0] | type=2
        u32x4 g0;
        g0[0] = 1u;                                              // count=1, is_restore=0
        g0[1] = lds;                                             // LDS byte address
        g0[2] = (uint32_t)ga;                                    // global_addr[31:0]
        g0[3] = ((uint32_t)(ga >> 32) & 0x01FFFFFFu) | (2u << 30); // addr[56:32] | type=2

        // D# group 1 (256b)
        u32x8 g1;
        g1[0] = 2u << 16;                  // wg_mask=0 | data_size=2 (4 bytes)
        g1[1] = ((uint32_t)HW_) << 16;     // tensor_dim0[15:0]=1024 (bits 63:48)
        g1[2] = ((uint32_t)C_)  << 16;     // tensor_dim0 hi=0 | tensor_dim1[15:0]=336
        g1[3] = ((uint32_t)TILE_) << 16;   // tensor_dim1 hi=0 | tile_dim0=64
        g1[4] = (uint32_t)rows;            // tile_dim1=168 | tile_dim2=0 (2D)
        g1[5] = (uint32_t)HW_;             // tensor_dim0_stride[31:0]=1024 elements
        g1[6] = 0u;                        // stride0 hi | tensor_dim1_stride lo (unused, 2D)
        g1[7] = 0u;

        asm volatile("tensor_load_to_lds %0, %1" :: "s"(g0), "s"(g1) : "memory");
        __builtin_amdgcn_s_wait_tensorcnt(0);   // wave's DMA complete before it signals barrier
    }
    __syncthreads();

    // ---------------- Fused decomposition, one HW position per thread ----------------
    const int   hw    = tid;
    const float inv_k = 1.0f / 25.0f;
    const float inv_g = 1.0f / 14.0f;

    // Phase accumulators for the seasonal component (static indices via unroll-by-24).
    float ph[PE_];
#pragma unroll
    for (int p = 0; p < PE_; ++p) ph[p] = 0.0f;

    // Window init for c=0: indices [-12..12] clamped -> 13*x[0] + x[1..12]
    float wsum = 13.0f * tile[hw];
#pragma unroll
    for (int j = 1; j <= 12; ++j) wsum += tile[j * TILE_ + hw];

    // Pass 1: sliding-window trend + phase sums of detrended signal
    for (int cb = 0; cb < C_; cb += PE_) {
#pragma unroll
        for (int p = 0; p < PE_; ++p) {
            const int   c  = cb + p;
            const float T  = wsum * inv_k;
            const float xd = tile[c * TILE_ + hw] - T;
            ph[p] += xd;
            int cin  = c + 13; if (cin  > C_ - 1) cin  = C_ - 1;
            int cout = c - 12; if (cout < 0)      cout = 0;
            wsum += tile[cin * TILE_ + hw] - tile[cout * TILE_ + hw];
        }
    }

    // Pass 2: re-run the cheap recurrence, emit T / S / R (coalesced streaming stores)
    float*       oT    = out;
    float*       oS    = out + OFFS_;
    float*       oR    = out + 2 * OFFS_;
    const size_t obase = (size_t)b * (C_ * HW_) + (size_t)t0 + (size_t)hw;

    wsum = 13.0f * tile[hw];
#pragma unroll
    for (int j = 1; j <= 12; ++j) wsum += tile[j * TILE_ + hw];

    for (int cb = 0; cb < C_; cb += PE_) {
#pragma unroll
        for (int p = 0; p < PE_; ++p) {
            const int    c  = cb + p;
            const float  T  = wsum * inv_k;
            const float  xd = tile[c * TILE_ + hw] - T;
            const float  S  = ph[p] * inv_g;
            const size_t o  = obase + (size_t)c * HW_;
            oT[o] = T;
            oS[o] = S;
            oR[o] = xd - S;
            int cin  = c + 13; if (cin  > C_ - 1) cin  = C_ - 1;
            int cout = c - 12; if (cout < 0)      cout = 0;
            wsum += tile[cin * TILE_ + hw] - tile[cout * TILE_ + hw];
        }
    }
}

extern "C" void kernel_launch(void* const* d_in, const int* in_sizes, int n_in,
                              void* d_out, int out_size, void* d_ws, size_t ws_size,
                              hipStream_t stream) {
    (void)in_sizes; (void)n_in; (void)out_size; (void)d_ws; (void)ws_size;
    const float* x   = (const float*)d_in[0];  // [32,336,32,32] f32
    float*       out = (float*)d_out;          // T | S | R concatenated

    const dim3 grid(B_ * (HW_ / TILE_));       // 512 workgroups
    const dim3 block(NTHR_);                   // 64 threads = 2 waves (wave32)
    decomp_tdm_kernel<<<grid, block, 0, stream>>>(x, out);
}